// RetDotProductMSR_69037304316600
// MI455X (gfx1250) — compile-verified
//
#include <hip/hip_runtime.h>
#include <hip/hip_bf16.h>
#include <stdint.h>

typedef __attribute__((ext_vector_type(2))) float v2f;
typedef __attribute__((ext_vector_type(8))) float v8f;

#define S_LEN 2048
#define DH 64
#define NH 16
#define NB 4
#define WAVES 8
#define ROWS_PER_WAVE 16
#define ROWS_PER_BLOCK (WAVES * ROWS_PER_WAVE) /* 128 */
#define TILEW 68  /* 64 + 4 pad: rows stay 16B-aligned, banks spread */
#define MSRW 17   /* 16 + 1 pad */

// Fused RetNet retention kernel:
//   scores = Q K^T / scale ; MSR = scores * D ; out = MSR @ V
// One wave handles a 16-row Q strip; matmuls via V_WMMA_F32_16X16X4_F32.
// K/V 16x64 tiles double-buffered via global_load_async_to_lds_b128 so the
// async DMA for block kb+16 overlaps the WMMA/D/MSR work on block kb.
__global__ __launch_bounds__(256) void retnet_msr_fused(
    const float* __restrict__ Q, const float* __restrict__ K,
    const float* __restrict__ V, const float* __restrict__ D,
    const int* __restrict__ scale_p,
    float* __restrict__ Out, float* __restrict__ MSR)
{
    __shared__ float ldsK[2][16][TILEW];               // K block double buffer
    __shared__ float ldsV[2][16][TILEW];               // V block double buffer
    __shared__ float ldsM[WAVES][ROWS_PER_WAVE][MSRW]; // per-wave MSR tile

    const int tid  = threadIdx.x;
    const int wave = tid >> 5;
    const int lane = tid & 31;
    const int hl   = lane >> 4;   // wave half (K/M split in frag layouts)
    const int m    = lane & 15;   // row (A) / column (B,C,D) index

    // h-major grid.y: the 4 batches sharing D[h] run adjacently -> L2 reuse of D.
    const int g  = blockIdx.y;    // 0 .. NB*NH-1
    const int h  = g >> 2;        // g / NB
    const int b  = g & (NB - 1);
    const int bh = b * NH + h;    // layout index into [B,H,...]

    const int qRow0 = blockIdx.x * ROWS_PER_BLOCK + wave * ROWS_PER_WAVE;

    const float inv_scale = 1.0f / (float)(*scale_p);

    const float* Qb = Q + ((size_t)bh * S_LEN + qRow0) * DH;
    const float* Kb = K + (size_t)bh * S_LEN * DH;
    const float* Vb = V + (size_t)bh * S_LEN * DH;
    const float* Db = D + (size_t)h  * S_LEN * S_LEN;
    float* MSRb = MSR + (size_t)bh * S_LEN * S_LEN;
    float* Outb = Out + ((size_t)bh * S_LEN + qRow0) * DH;

    // Cooperative staging slot for this thread: row 0..15, 16B chunk 0..15.
    const int sRowIdx = tid >> 4;   // 0..15
    const int sChunk  = tid & 15;   // 0..15 (x 4 floats)
    unsigned ldsKAddr[2], ldsVAddr[2];
    ldsKAddr[0] = (unsigned)(uintptr_t)&ldsK[0][sRowIdx][sChunk * 4];
    ldsKAddr[1] = (unsigned)(uintptr_t)&ldsK[1][sRowIdx][sChunk * 4];
    ldsVAddr[0] = (unsigned)(uintptr_t)&ldsV[0][sRowIdx][sChunk * 4];
    ldsVAddr[1] = (unsigned)(uintptr_t)&ldsV[1][sRowIdx][sChunk * 4];
    const float* gKrow = Kb + (size_t)sRowIdx * DH + sChunk * 4;
    const float* gVrow = Vb + (size_t)sRowIdx * DH + sChunk * 4;

    // One async b128 per thread per tile (16 rows x 64 floats each).
    auto stage = [&](int buf, int kb) {
        const uint64_t gK = (uint64_t)(uintptr_t)(gKrow + (size_t)kb * DH);
        const uint64_t gV = (uint64_t)(uintptr_t)(gVrow + (size_t)kb * DH);
        asm volatile("global_load_async_to_lds_b128 %0, %1, off"
                     :: "v"(ldsKAddr[buf]), "v"(gK) : "memory");
        asm volatile("global_load_async_to_lds_b128 %0, %1, off"
                     :: "v"(ldsVAddr[buf]), "v"(gV) : "memory");
    };

    // ---- Preload Q strip as 16 A-fragments (16x4 f32 layout).
    // Lanes 0-15: row m, K = 4f+{0,1}; lanes 16-31: row m, K = 4f+{2,3}.
    v2f qA[16];
#pragma unroll
    for (int f = 0; f < 16; ++f) {
        qA[f] = __builtin_nontemporal_load(
                    (const v2f*)(Qb + (size_t)m * DH + 4 * f + 2 * hl));
    }

    // Output accumulators: 4 tiles of 16x16 covering DH=64 columns.
    v8f outAcc[4];
#pragma unroll
    for (int t = 0; t < 4; ++t) outAcc[t] = (v8f){};

    stage(0, 0);   // prologue: stage first K/V block

    for (int kb = 0; kb < S_LEN; kb += 16) {
        const int cur = (kb >> 4) & 1;

        // Wait for this iteration's tiles (issued last iteration), make them
        // visible to all waves, then immediately kick off the next block's DMA.
        asm volatile("s_wait_asynccnt 0" ::: "memory");
        __syncthreads();
        if (kb + 16 < S_LEN) stage(cur ^ 1, kb + 16);

        // ---- GEMM1: 16x16 score tile = Q_strip (16xDH) x K_block^T (DHx16)
        // B fragment: B[k][n] = K[kb+n][k]; lane n=m -> float2 from ldsK row m.
        v8f c = (v8f){};
#pragma unroll
        for (int f = 0; f < 16; ++f) {
            v2f bK = *(const v2f*)&ldsK[cur][m][4 * f + 2 * hl];
            c = __builtin_amdgcn_wmma_f32_16x16x4_f32(
                    false, qA[f], false, bK, (short)0, c, false, false);
        }

        // ---- Elementwise: *1/scale, *D; MSR -> global (NT) + own LDS tile.
        // C layout: VGPR r -> M = r + 8*hl, N = m.
        const float* dRow = Db   + (size_t)(qRow0 + 8 * hl) * S_LEN + kb + m;
        float*       sRow = MSRb + (size_t)(qRow0 + 8 * hl) * S_LEN + kb + m;
#pragma unroll
        for (int r = 0; r < 8; ++r) {
            float v = c[r] * inv_scale * dRow[(size_t)r * S_LEN];
            __builtin_nontemporal_store(v, sRow + (size_t)r * S_LEN);
            ldsM[wave][8 * hl + r][m] = v;
        }

        // ---- GEMM2: outAcc(16x64) += MSR_tile(16x16) @ V_block(16x64)
        // (ldsM is wave-private; LDS ops are in-order within a wave. All ds
        // reads below are consumed by WMMAs this iteration, so each wave's
        // DScnt drains before it reaches the next barrier -> safe to overwrite
        // the other buffer after that barrier.)
#pragma unroll
        for (int f = 0; f < 4; ++f) {
            v2f aM;
            aM.x = ldsM[wave][m][4 * f + 2 * hl + 0];
            aM.y = ldsM[wave][m][4 * f + 2 * hl + 1];
            const int k0 = 4 * f + 2 * hl;   // row within staged V tile
#pragma unroll
            for (int t = 0; t < 4; ++t) {
                v2f bV;
                bV.x = ldsV[cur][k0 + 0][t * 16 + m];
                bV.y = ldsV[cur][k0 + 1][t * 16 + m];
                outAcc[t] = __builtin_amdgcn_wmma_f32_16x16x4_f32(
                                false, aM, false, bV, (short)0, outAcc[t], false, false);
            }
        }
    }

    // ---- Store 16x64 output strip (C layout: M = r + 8*hl, N = m), streaming.
#pragma unroll
    for (int t = 0; t < 4; ++t) {
#pragma unroll
        for (int r = 0; r < 8; ++r) {
            __builtin_nontemporal_store(
                outAcc[t][r], Outb + (size_t)(8 * hl + r) * DH + t * 16 + m);
        }
    }
}

extern "C" void kernel_launch(void* const* d_in, const int* in_sizes, int n_in,
                              void* d_out, int out_size, void* d_ws, size_t ws_size,
                              hipStream_t stream) {
    const float* Q = (const float*)d_in[0];
    const float* K = (const float*)d_in[1];
    const float* V = (const float*)d_in[2];
    const float* D = (const float*)d_in[3];
    const int* scale = (const int*)d_in[4];

    float* Out = (float*)d_out;                               // [B,H,S,DH]
    float* MSR = Out + (size_t)NB * NH * S_LEN * DH;          // [B,H,S,S]

    dim3 grid(S_LEN / ROWS_PER_BLOCK, NB * NH);
    retnet_msr_fused<<<grid, 256, 0, stream>>>(Q, K, V, D, scale, Out, MSR);
}